// SAGE_69002944578218
// MI455X (gfx1250) — compile-verified
//
#include <hip/hip_runtime.h>
#include <stdint.h>

#define N_NODES 100000
#define F_IN 128
#define F_H 128
#define F_OUT 64
#define BN_EPS 1e-5f

typedef __attribute__((ext_vector_type(2))) float v2f;
typedef __attribute__((ext_vector_type(8))) float v8f;

// ---------------- degree: atomicAdd 1.0 per edge ----------------
__global__ void sage_degree_kernel(const int* __restrict__ dst, int E,
                                   float* __restrict__ deg) {
  int e = blockIdx.x * blockDim.x + threadIdx.x;
  if (e < E) atomicAdd(&deg[dst[e]], 1.0f);
}

// ---------------- edge scatter: one wave per edge, float4 per lane -------
// feat: [N,128] fp32, acc: [N,128] fp32 (pre-zeroed)
__global__ void sage_scatter_kernel(const float* __restrict__ feat,
                                    const int* __restrict__ src,
                                    const int* __restrict__ dst, int E,
                                    float* __restrict__ acc) {
  long long gid = (long long)blockIdx.x * blockDim.x + threadIdx.x;
  int e = (int)(gid >> 5);
  int lane = (int)(gid & 31);
  if (e >= E) return;
  int s = src[e];
  int d = dst[e];
  float4 v = reinterpret_cast<const float4*>(feat + (size_t)s * F_IN)[lane];
  float* out = acc + (size_t)d * F_IN + lane * 4;
  atomicAdd(out + 0, v.x);
  atomicAdd(out + 1, v.y);
  atomicAdd(out + 2, v.z);
  atomicAdd(out + 3, v.w);
}

// ---------------- layer 1 GEMM: h = relu(bn(agg/deg @ Wl^T + x @ Wr^T + bl))
// Block = 32 node rows (2 row tiles). LDS panel As[32][256] = [agg_norm | x],
// stride 257. 8 waves; wave w owns output cols [16w,16w+16) with TWO
// accumulators (rows 0-15 and 16-31) sharing each B fragment.
__global__ __launch_bounds__(256) void sage_gemm1_kernel(
    const float* __restrict__ agg, const float* __restrict__ deg,
    const float* __restrict__ x, const float* __restrict__ Wl,
    const float* __restrict__ bl, const float* __restrict__ Wr,
    const float* __restrict__ bn_g, const float* __restrict__ bn_b,
    const float* __restrict__ bn_rm, const float* __restrict__ bn_rv,
    float* __restrict__ h) {
  __shared__ float As[32 * 257];
  const int tid = threadIdx.x;
  const int rowbase = blockIdx.x * 32;

  // stage A panel: 32 rows x 32 float4 of agg_norm and of x
#pragma unroll
  for (int it = 0; it < 4; ++it) {
    int slot = tid + it * 256;      // 0..1023
    int r = slot >> 5;              // 0..31
    int c4s = slot & 31;            // float4 slot 0..31
    int node = rowbase + r;
    float inv = 1.0f / fmaxf(deg[node], 1.0f);
    float4 a = reinterpret_cast<const float4*>(agg + (size_t)node * F_IN)[c4s];
    float4 xv = reinterpret_cast<const float4*>(x + (size_t)node * F_IN)[c4s];
    int c = c4s * 4;
    float* rowp = &As[r * 257];
    rowp[c + 0] = a.x * inv;  rowp[c + 1] = a.y * inv;
    rowp[c + 2] = a.z * inv;  rowp[c + 3] = a.w * inv;
    rowp[128 + c + 0] = xv.x; rowp[128 + c + 1] = xv.y;
    rowp[128 + c + 2] = xv.z; rowp[128 + c + 3] = xv.w;
  }
  __syncthreads();

  const int wave = tid >> 5;
  const int lane = tid & 31;
  const int n0 = wave * 16;
  const int m = lane & 15;
  const int khalf = (lane >> 4) * 2;   // lanes 16-31 hold K+2,K+3
  const int n = n0 + m;

  v8f c0 = {};
  v8f c1 = {};
#pragma unroll 4
  for (int k = 0; k < 256; k += 4) {
    int kk = k + khalf;
    v2f a0, a1;
    a0.x = As[m * 257 + kk];
    a0.y = As[m * 257 + kk + 1];
    a1.x = As[(16 + m) * 257 + kk];
    a1.y = As[(16 + m) * 257 + kk + 1];
    const float* W = (k < 128) ? Wl : Wr;
    int kw = (k < 128) ? kk : (kk - 128);
    float2 bw = *reinterpret_cast<const float2*>(W + (size_t)n * F_IN + kw);
    v2f b;
    b.x = bw.x;
    b.y = bw.y;
    c0 = __builtin_amdgcn_wmma_f32_16x16x4_f32(false, a0, false, b, (short)0,
                                               c0, false, false);
    c1 = __builtin_amdgcn_wmma_f32_16x16x4_f32(false, a1, false, b, (short)0,
                                               c1, false, false);
  }

  // fused bias + BN(eval) + ReLU, per output column n
  float scale = bn_g[n] * rsqrtf(bn_rv[n] + BN_EPS);
  float off = bl[n] - bn_rm[n];
  float shift = bn_b[n];
  const int hi8 = (lane >> 4) * 8;    // C layout: VGPR i -> M=i / i+8
#pragma unroll
  for (int i = 0; i < 8; ++i) {
    int row0 = rowbase + i + hi8;
    int row1 = rowbase + 16 + i + hi8;
    float v0 = fmaxf((c0[i] + off) * scale + shift, 0.0f);
    float v1 = fmaxf((c1[i] + off) * scale + shift, 0.0f);
    h[(size_t)row0 * F_H + n] = v0;
    h[(size_t)row1 * F_H + n] = v1;
  }
}

// ---------------- layer 2 GEMM: out = agg/deg @ Wl2^T + bl2 + h @ Wr2^T ----
// Block = 32 node rows, 128 threads = 4 waves. Wave w owns output cols
// [16w,16w+16) of OUT=64 with TWO row-tile accumulators sharing B.
__global__ __launch_bounds__(128) void sage_gemm2_kernel(
    const float* __restrict__ agg, const float* __restrict__ deg,
    const float* __restrict__ h, const float* __restrict__ Wl,
    const float* __restrict__ bl, const float* __restrict__ Wr,
    float* __restrict__ out) {
  __shared__ float As[32 * 257];
  const int tid = threadIdx.x;
  const int rowbase = blockIdx.x * 32;

#pragma unroll
  for (int it = 0; it < 8; ++it) {
    int slot = tid + it * 128;      // 0..1023
    int r = slot >> 5;              // 0..31
    int c4s = slot & 31;
    int node = rowbase + r;
    float inv = 1.0f / fmaxf(deg[node], 1.0f);
    float4 a = reinterpret_cast<const float4*>(agg + (size_t)node * F_H)[c4s];
    float4 hv = reinterpret_cast<const float4*>(h + (size_t)node * F_H)[c4s];
    int c = c4s * 4;
    float* rowp = &As[r * 257];
    rowp[c + 0] = a.x * inv;  rowp[c + 1] = a.y * inv;
    rowp[c + 2] = a.z * inv;  rowp[c + 3] = a.w * inv;
    rowp[128 + c + 0] = hv.x; rowp[128 + c + 1] = hv.y;
    rowp[128 + c + 2] = hv.z; rowp[128 + c + 3] = hv.w;
  }
  __syncthreads();

  const int wave = tid >> 5;         // 0..3 -> col tile
  const int lane = tid & 31;
  const int n0 = wave * 16;
  const int m = lane & 15;
  const int khalf = (lane >> 4) * 2;
  const int n = n0 + m;

  v8f c0 = {};
  v8f c1 = {};
#pragma unroll 4
  for (int k = 0; k < 256; k += 4) {
    int kk = k + khalf;
    v2f a0, a1;
    a0.x = As[m * 257 + kk];
    a0.y = As[m * 257 + kk + 1];
    a1.x = As[(16 + m) * 257 + kk];
    a1.y = As[(16 + m) * 257 + kk + 1];
    const float* W = (k < 128) ? Wl : Wr;
    int kw = (k < 128) ? kk : (kk - 128);
    float2 bw = *reinterpret_cast<const float2*>(W + (size_t)n * F_H + kw);
    v2f b;
    b.x = bw.x;
    b.y = bw.y;
    c0 = __builtin_amdgcn_wmma_f32_16x16x4_f32(false, a0, false, b, (short)0,
                                               c0, false, false);
    c1 = __builtin_amdgcn_wmma_f32_16x16x4_f32(false, a1, false, b, (short)0,
                                               c1, false, false);
  }

  float bias = bl[n];
  const int hi8 = (lane >> 4) * 8;
#pragma unroll
  for (int i = 0; i < 8; ++i) {
    int row0 = rowbase + i + hi8;
    int row1 = rowbase + 16 + i + hi8;
    out[(size_t)row0 * F_OUT + n] = c0[i] + bias;
    out[(size_t)row1 * F_OUT + n] = c1[i] + bias;
  }
}

extern "C" void kernel_launch(void* const* d_in, const int* in_sizes, int n_in,
                              void* d_out, int out_size, void* d_ws,
                              size_t ws_size, hipStream_t stream) {
  const float* x    = (const float*)d_in[0];
  const int*   ei   = (const int*)d_in[1];   // [2,E]: row0=src, row1=dst
  const float* Wl1  = (const float*)d_in[2];
  const float* bl1  = (const float*)d_in[3];
  const float* Wr1  = (const float*)d_in[4];
  const float* Wl2  = (const float*)d_in[5];
  const float* bl2  = (const float*)d_in[6];
  const float* Wr2  = (const float*)d_in[7];
  const float* bn_g = (const float*)d_in[8];
  const float* bn_b = (const float*)d_in[9];
  const float* bn_rm = (const float*)d_in[10];
  const float* bn_rv = (const float*)d_in[11];

  const int E = in_sizes[1] / 2;
  const int* src = ei;
  const int* dst = ei + E;

  // workspace layout: deg[N] | agg[N*128] | h[N*128]
  char* ws = (char*)d_ws;
  float* deg = (float*)ws;
  size_t off = ((size_t)N_NODES * 4 + 255) & ~(size_t)255;
  float* agg = (float*)(ws + off);
  off += (size_t)N_NODES * F_IN * 4;
  float* hbuf = (float*)(ws + off);

  hipMemsetAsync(deg, 0, (size_t)N_NODES * 4, stream);
  hipMemsetAsync(agg, 0, (size_t)N_NODES * F_IN * 4, stream);

  sage_degree_kernel<<<(E + 255) / 256, 256, 0, stream>>>(dst, E, deg);

  unsigned scatter_blocks = (unsigned)(((long long)E * 32 + 255) / 256);
  sage_scatter_kernel<<<scatter_blocks, 256, 0, stream>>>(x, src, dst, E, agg);

  sage_gemm1_kernel<<<N_NODES / 32, 256, 0, stream>>>(
      agg, deg, x, Wl1, bl1, Wr1, bn_g, bn_b, bn_rm, bn_rv, hbuf);

  hipMemsetAsync(agg, 0, (size_t)N_NODES * F_IN * 4, stream);

  sage_scatter_kernel<<<scatter_blocks, 256, 0, stream>>>(hbuf, src, dst, E,
                                                          agg);

  sage_gemm2_kernel<<<N_NODES / 32, 128, 0, stream>>>(agg, deg, hbuf, Wl2, bl2,
                                                      Wr2, (float*)d_out);
}